// NMS_MODEL_38070590111793
// MI455X (gfx1250) — compile-verified
//
#include <hip/hip_runtime.h>
#include <math.h>

typedef __attribute__((ext_vector_type(2))) float v2f;
typedef __attribute__((ext_vector_type(8))) float v8f;

static constexpr int Bn = 4, Hn = 1024, Wn = 1024;
static constexpr int NPIX = Bn * Hn * Wn;

// ---------- order-preserving float<->uint keys for deterministic min/max atomics ----------
__device__ __forceinline__ unsigned f2key(float f) {
  unsigned u = __float_as_uint(f);
  return (u & 0x80000000u) ? ~u : (u | 0x80000000u);
}
__device__ __forceinline__ float key2f(unsigned k) {
  unsigned u = (k & 0x80000000u) ? (k ^ 0x80000000u) : ~k;
  return __uint_as_float(u);
}

// ---------------- K1: E = separable 3-tap smooth of img (zero padded) ----------------
__global__ __launch_bounds__(256) void k_smooth3(const float* __restrict__ img,
                                                 const float* __restrict__ g3,
                                                 float* __restrict__ E) {
  int idx = blockIdx.x * 256 + threadIdx.x;
  int x = idx & (Wn - 1);
  int y = (idx >> 10) & (Hn - 1);
  int b = idx >> 20;
  const float* im = img + (long)b * Hn * Wn;
  float cf[3] = {g3[0], g3[1], g3[2]};
  float acc = 0.f;
#pragma unroll
  for (int i = 0; i < 3; ++i) {
    int yy = y + i - 1;
    if ((unsigned)yy >= (unsigned)Hn) continue;
    float r = 0.f;
#pragma unroll
    for (int j = 0; j < 3; ++j) {
      int xx = x + j - 1;
      if ((unsigned)xx >= (unsigned)Wn) continue;
      r += cf[j] * im[yy * Wn + xx];
    }
    acc += cf[i] * r;
  }
  E[idx] = acc;
}

// ---------------- K2/K3: 9-tap conv along contiguous dim, transposed output ----------
// in: 1024 x 1024 per image. out[q][p] = sum_t g9[t]*in[p][q+t-4] (zero pad in q),
// stored transposed (out is 1024 x 1024 with roles swapped).
// One wave per 16x16 output tile: D(16x16) = A(16x24 banded coeff) x B(24x16 input),
// chained as 6 x V_WMMA_F32_16X16X4_F32.
#if __has_builtin(__builtin_amdgcn_wmma_f32_16x16x4_f32)
#define USE_WMMA_F32 1
#else
#define USE_WMMA_F32 0
#endif

static constexpr int PITCH = 28;  // 28*m mod 64 distinct for m=0..15; 16B aligned rows

__global__ __launch_bounds__(256) void k_conv9_tr(const float* __restrict__ in,
                                                  const float* __restrict__ g9,
                                                  float* __restrict__ out) {
  __shared__ __align__(16) float lds[8][16 * PITCH];
  __shared__ float gext[40];  // zero-extended band table: g9 at [16..24], 0 elsewhere
  const int lane = threadIdx.x & 31;
  const int wid = threadIdx.x >> 5;

  if (threadIdx.x < 40) {
    int i = threadIdx.x;
    gext[i] = (i >= 16 && i < 25) ? g9[i - 16] : 0.f;
  }

  int tile = blockIdx.x * 8 + wid;         // 4 images * 64 * 64 tiles
  int b = tile >> 12;
  int t = tile & 4095;
  int q0 = (t >> 6) << 4;                  // output-row block (conv dim)
  int p0 = (t & 63) << 4;                  // output-col block
  const float* ip = in + ((long)b << 20);
  float* op = out + ((long)b << 20);

  // Stage 16x24 input tile (rows p0.., cols q0-4..q0+19) as guarded float4 loads.
  float* T = lds[wid];
#pragma unroll
  for (int i = 0; i < 3; ++i) {
    int seg = i * 32 + lane;               // 96 float4 segments
    int r = seg / 6;
    int c4 = seg - r * 6;
    int qc = q0 - 4 + c4 * 4;              // 4-aligned; fully in or out of [0,1024)
    float4 v = make_float4(0.f, 0.f, 0.f, 0.f);
    if ((unsigned)qc < 1024u) v = *(const float4*)(ip + ((p0 + r) << 10) + qc);
    *(float4*)(T + r * PITCH + c4 * 4) = v;
  }
  __syncthreads();

  const int m = lane & 15;  // A row / B col / D col
  const int h = lane >> 4;  // K half (f32 layout: low lanes K=0,1; high lanes K=2,3)

  // Gather all operands first (LDS only, branch-free), then burst the WMMA chain.
  float av[12], bv[12];
#pragma unroll
  for (int cb = 0; cb < 6; ++cb) {
    int j0 = cb * 4 + 2 * h;
    av[2 * cb]     = gext[j0 - m + 16];    // A[m][j0]   = g9[j0-m] or 0
    av[2 * cb + 1] = gext[j0 - m + 17];    // A[m][j0+1]
    bv[2 * cb]     = T[m * PITCH + j0];    // B[j0][m]
    bv[2 * cb + 1] = T[m * PITCH + j0 + 1];
  }
  v8f acc = {};
#pragma unroll
  for (int cb = 0; cb < 6; ++cb) {
#if USE_WMMA_F32
    v2f a = {av[2 * cb], av[2 * cb + 1]};
    v2f bb = {bv[2 * cb], bv[2 * cb + 1]};
    acc = __builtin_amdgcn_wmma_f32_16x16x4_f32(false, a, false, bb, (short)0, acc,
                                                false, false);
#endif
  }
#if !USE_WMMA_F32
  // mathematically identical fallback (lane computes its 8 D elements directly)
#pragma unroll
  for (int j = 0; j < 8; ++j) {
    float s = 0.f;
#pragma unroll
    for (int tt = 0; tt < 9; ++tt) s += g9[tt] * T[m * PITCH + j + 8 * h + tt];
    acc[j] = s;
  }
#endif
  // D VGPR j holds rows q0+j (lanes 0-15) and q0+j+8 (lanes 16-31), col = p0+m.
#pragma unroll
  for (int j = 0; j < 8; ++j) {
    int q = q0 + j + 8 * h;
    op[((long)q << 10) + p0 + m] = acc[j];
  }
}

// ---------------- K4: orientation from E1 (cascaded Sobels w/ intermediate zero-pad) ---
__global__ __launch_bounds__(256) void k_orient(const float* __restrict__ E1,
                                                const float* __restrict__ sb,
                                                float* __restrict__ O) {
  int idx = blockIdx.x * 256 + threadIdx.x;
  int x = idx & (Wn - 1);
  int y = (idx >> 10) & (Hn - 1);
  int b = idx >> 20;
  const float* p = E1 + (long)b * Hn * Wn;
  float sk[9];
#pragma unroll
  for (int i = 0; i < 9; ++i) sk[i] = sb[i];

  float n[5][5];
#pragma unroll
  for (int a = 0; a < 5; ++a)
#pragma unroll
    for (int c = 0; c < 5; ++c) {
      int yy = y + a - 2, xx = x + c - 2;
      n[a][c] = ((unsigned)yy < (unsigned)Hn && (unsigned)xx < (unsigned)Wn)
                    ? p[yy * Wn + xx] : 0.f;
    }
  float gx[3][3], gy[3][3];
#pragma unroll
  for (int di = -1; di <= 1; ++di)
#pragma unroll
    for (int dj = -1; dj <= 1; ++dj) {
      bool inside = ((unsigned)(y + di) < (unsigned)Hn) && ((unsigned)(x + dj) < (unsigned)Wn);
      float sx = 0.f, sy = 0.f;
#pragma unroll
      for (int i = 0; i < 3; ++i)
#pragma unroll
        for (int j = 0; j < 3; ++j) {
          float v = n[di + i + 1][dj + j + 1];
          sx += sk[i * 3 + j] * v;  // sbh
          sy += sk[j * 3 + i] * v;  // sbv = sbh^T
        }
      gx[di + 1][dj + 1] = inside ? sx : 0.f;
      gy[di + 1][dj + 1] = inside ? sy : 0.f;
    }
  float gxx = 0.f, gxy = 0.f, gyy = 0.f;
#pragma unroll
  for (int i = 0; i < 3; ++i)
#pragma unroll
    for (int j = 0; j < 3; ++j) {
      gxx += sk[i * 3 + j] * gx[i][j];  // sobel_h(grad_x)
      gxy += sk[j * 3 + i] * gx[i][j];  // sobel_v(grad_x)
      gyy += sk[j * 3 + i] * gy[i][j];  // sobel_v(grad_y)
    }
  float ngxy = -gxy;
  float s = (ngxy > 0.f) ? 1.f : ((ngxy < 0.f) ? -1.f : 0.f);
  float a = atanf(gyy * s / (gxx + 1e-5f));
  O[idx] = a - 3.14f * floorf(a / 3.14f);  // jnp.mod semantics
}

// ---------------- K5: directional NMS via bilinear taps + border down-weight ----------
__global__ __launch_bounds__(256) void k_nms(const float* __restrict__ E,
                                             const float* __restrict__ O,
                                             float* __restrict__ E0) {
  int idx = blockIdx.x * 256 + threadIdx.x;
  int x = idx & (Wn - 1);
  int y = (idx >> 10) & (Hn - 1);
  int b = idx >> 20;
  const float* e = E + (long)b * Hn * Wn;
  float cur = 1.01f * e[y * Wn + x];
  float o = O[idx];
  float co = cosf(o), si = sinf(o);
  float basex = -1.f + (float)x * (2.f / (float)(Wn - 1));
  float basey = -1.f + (float)y * (2.f / (float)(Hn - 1));

  auto tap = [&](float xf, float yf) -> float {
    bool valid = (xf >= 0.f) && (xf <= (float)(Wn - 1)) && (yf >= 0.f) && (yf <= (float)(Hn - 1));
    int xi = (int)fminf(fmaxf(xf, 0.f), (float)(Wn - 1));
    int yi = (int)fminf(fmaxf(yf, 0.f), (float)(Hn - 1));
    float v = e[yi * Wn + xi];
    return valid ? v : 0.f;
  };

  const int dlist[4] = {-2, -1, 1, 2};
#pragma unroll
  for (int k = 0; k < 4; ++k) {
    float d = (float)dlist[k];
    float gx1 = basex - co * (d / (float)Wn);
    float gy1 = basey - si * (d / (float)Hn);
    float px = (gx1 + 1.f) * (0.5f * (float)(Wn - 1));
    float py = (gy1 + 1.f) * (0.5f * (float)(Hn - 1));
    float x0 = floorf(px), y0 = floorf(py);
    float wx1 = px - x0, wx0 = 1.f - wx1;
    float wy1 = py - y0, wy0 = 1.f - wy1;
    float nb = tap(x0, y0) * (wx0 * wy0) + tap(x0 + 1.f, y0) * (wx1 * wy0) +
               tap(x0, y0 + 1.f) * (wx0 * wy1) + tap(x0 + 1.f, y0 + 1.f) * (wx1 * wy1);
    nb *= 1.01f;
    if (nb > cur) cur = 0.f;
  }
  float f = 1.f;
  if (x < 5) f *= 0.2f;
  if (x >= Wn - 6) f *= 0.2f;
  if (y < 5) f *= 0.2f;
  if (y >= Hn - 6) f *= 0.2f;
  E0[idx] = cur * f;
}

// ---------------- K6: ring-sign gating + deterministic global min/max ----------------
__global__ __launch_bounds__(256) void k_ring(const float* __restrict__ E0,
                                              float* __restrict__ E0b,
                                              unsigned* __restrict__ mmk) {
  int idx = blockIdx.x * 256 + threadIdx.x;
  int x = idx & (Wn - 1);
  int y = (idx >> 10) & (Hn - 1);
  int b = idx >> 20;
  const float* p = E0 + (long)b * Hn * Wn;
  float s5 = 0.f, s3 = 0.f;
#pragma unroll
  for (int a = -2; a <= 2; ++a)
#pragma unroll
    for (int c = -2; c <= 2; ++c) {
      bool on5 = (a == -2 || a == 2 || c == -2 || c == 2);
      bool on3 = (!on5) && !(a == 0 && c == 0);
      if (!on5 && !on3) continue;
      int yy = y + a, xx = x + c;
      if ((unsigned)yy >= (unsigned)Hn || (unsigned)xx >= (unsigned)Wn) continue;
      float v = p[yy * Wn + xx];
      if (on5) s5 += v; else s3 += v;
    }
  float sg5 = (s5 > 0.f) ? 1.f : ((s5 < 0.f) ? -1.f : 0.f);
  float sg3 = (s3 > 0.f) ? 1.f : ((s3 < 0.f) ? -1.f : 0.f);
  float r = p[y * Wn + x] * sg5 * sg3;
  E0b[idx] = r;

  __shared__ unsigned smn[256], smx[256];
  unsigned k = f2key(r);
  int tid = threadIdx.x;
  smn[tid] = k;
  smx[tid] = k;
  __syncthreads();
#pragma unroll
  for (int s = 128; s > 0; s >>= 1) {
    if (tid < s) {
      smn[tid] = min(smn[tid], smn[tid + s]);
      smx[tid] = max(smx[tid], smx[tid + s]);
    }
    __syncthreads();
  }
  if (tid == 0) {
    atomicMin(&mmk[0], smn[0]);
    atomicMax(&mmk[1], smx[0]);
  }
}

__global__ void k_init(unsigned* mmk) {
  mmk[0] = 0xFFFFFFFFu;
  mmk[1] = 0u;
}

// ---------------- K7: normalize + 3-way threshold ----------------
__global__ __launch_bounds__(256) void k_final(const float* __restrict__ E0b,
                                               const unsigned* __restrict__ mmk,
                                               float* __restrict__ out) {
  int idx = blockIdx.x * 256 + threadIdx.x;
  float mn = key2f(mmk[0]);
  float mx = key2f(mmk[1]);
  float nrm = (E0b[idx] - mn) / (mx - mn);
  out[idx] = (nrm > 0.5f) ? 1.f : ((nrm < 0.5f) ? 0.f : 2.f);
}

// -------------------------------------------------------------------------------------
extern "C" void kernel_launch(void* const* d_in, const int* in_sizes, int n_in,
                              void* d_out, int out_size, void* d_ws, size_t ws_size,
                              hipStream_t stream) {
  (void)in_sizes; (void)n_in; (void)out_size; (void)ws_size;
  const float* img = (const float*)d_in[0];
  const float* gh1 = (const float*)d_in[1];  // 3-tap gaussian
  const float* gfh = (const float*)d_in[3];  // 9-tap gaussian
  const float* sbh = (const float*)d_in[5];  // sobel 3x3 (row-major)
  float* out = (float*)d_out;

  float* b0 = (float*)d_ws;          // E, later E0b
  float* b1 = b0 + NPIX;             // T (transposed intermediate), later O
  float* b2 = b1 + NPIX;             // E1, later E0
  unsigned* mmk = (unsigned*)(b2 + NPIX);

  dim3 blk(256);
  dim3 grid(NPIX / 256);
  const int tiles = Bn * (Hn / 16) * (Wn / 16);  // 16384 tiles, 8 waves/block

  // E = smooth3(img)
  k_smooth3<<<grid, blk, 0, stream>>>(img, gh1, b0);
  // T = conv9 along W of E, stored transposed (W x H per image)
  k_conv9_tr<<<tiles / 8, blk, 0, stream>>>(b0, gfh, b1);
  // E1 = conv9 along H of T (contiguous dim of transposed layout), stored back H x W
  k_conv9_tr<<<tiles / 8, blk, 0, stream>>>(b1, gfh, b2);
  // O = orientation(E1)
  k_orient<<<grid, blk, 0, stream>>>(b2, sbh, b1);
  // E0 = NMS(E, O) + border scale
  k_nms<<<grid, blk, 0, stream>>>(b0, b1, b2);
  // min/max init + ring-sign gating with reduction
  k_init<<<1, 1, 0, stream>>>(mmk);
  k_ring<<<grid, blk, 0, stream>>>(b2, b0, mmk);
  // normalize + threshold
  k_final<<<grid, blk, 0, stream>>>(b0, mmk, out);
}